// LSTMModel_88605175317034
// MI455X (gfx1250) — compile-verified
//
#include <hip/hip_runtime.h>

// ---------------------------------------------------------------------------
// LSTM (2 layers, B=512, T=512, D=64, H=128) + MLP head for MI455X (gfx1250).
// fp32 WMMA (V_WMMA_F32_16X16X4_F32) keeps reference precision; recurrence is
// parallelized over independent batch rows with w_hh resident in 320KB LDS.
// Weight staging uses CDNA5 async global->LDS (ASYNCcnt); the recurrence xz
// loads are register software-pipelined one step ahead.
// ---------------------------------------------------------------------------

typedef float v2f __attribute__((ext_vector_type(2)));
typedef float v8f __attribute__((ext_vector_type(8)));

#define WMMA_F32(a, b, c) \
  __builtin_amdgcn_wmma_f32_16x16x4_f32(false, (a), false, (b), (short)0, (c), false, false)

constexpr int Bsz = 512;   // batch
constexpr int Tn  = 512;   // timesteps
constexpr int Dn  = 64;    // input features
constexpr int Hn  = 128;   // hidden
constexpr int G4  = 512;   // 4*H (gate width)

__device__ __forceinline__ float sig_(float x) {
  return __builtin_amdgcn_rcpf(1.0f + __expf(-x));
}
__device__ __forceinline__ float tanh_(float x) {
  return 1.0f - 2.0f * __builtin_amdgcn_rcpf(1.0f + __expf(2.0f * x));
}

// CDNA5 async global->LDS copy (16B per lane), tracked by ASYNCcnt.
// lds_off = low 32 bits of a shared-space pointer (= LDS byte offset).
__device__ __forceinline__ void async_g2l_b128(unsigned lds_off, const float* g) {
  asm volatile("global_load_async_to_lds_b128 %0, %1, off"
               :: "v"(lds_off), "v"(g) : "memory");
}
__device__ __forceinline__ void wait_async0() {
  asm volatile("s_wait_asynccnt 0x0" ::: "memory");
}

// ---------------------------------------------------------------------------
// xz_gemm: Y[r][0..511] = Xrow(r) . W[j][:]  + bias1[j] + bias2[j]
//   r = t*512 + b ; Xrow(r) = X + (r&511)*strideB + (r>>9)*strideT
//   Layer 0: X = x [B][T][D]    (strideB=T*D, strideT=D,   K=64)
//   Layer 1: X = out0 [T][B][H] (strideB=H,   strideT=B*H, K=128)
// Workgroup: 256 thr = 8 waves, tile = 128 rows x 128 cols; wave = 16 rows.
// W slice [128 rows][K] staged to LDS via async copies, padded pitch.
// ---------------------------------------------------------------------------
__global__ __launch_bounds__(256) void xz_gemm(
    const float* __restrict__ X, int strideB, int strideT,
    const float* __restrict__ W, int K, int kshift, int pitch,
    const float* __restrict__ bias1, const float* __restrict__ bias2,
    float* __restrict__ Y)
{
  extern __shared__ float ldsW[];  // [128][pitch]
  const int tid  = threadIdx.x;
  const int wave = tid >> 5;
  const int lane = tid & 31;
  const int col0 = blockIdx.y * 128;

  // async-stage the 128-row W slice (16B per lane per op)
  const int kq4 = K >> 2;
  for (int i4 = tid; i4 < 128 * kq4; i4 += 256) {
    const int jj = i4 >> (kshift - 2);
    const int kq = (i4 & (kq4 - 1)) << 2;
    async_g2l_b128((unsigned)(uintptr_t)&ldsW[jj * pitch + kq],
                   W + (size_t)(col0 + jj) * K + kq);
  }
  wait_async0();
  __syncthreads();

  const int ln   = lane & 15;
  const int lh   = lane >> 4;      // half-wave select (K offset +2 / M offset +8)
  const int kofs = 2 * lh;
  const int r0   = blockIdx.x * 128 + wave * 16;
  const int rA   = r0 + ln;        // this lane's A-matrix row
  const float* Arow = X + (size_t)(rA & 511) * strideB
                        + (size_t)(rA >> 9) * strideT + kofs;

  v8f acc[8];
  #pragma unroll
  for (int j = 0; j < 8; ++j) acc[j] = (v8f)(0.0f);

  for (int kb = 0; kb < K; kb += 4) {
    const v2f a = *(const v2f*)(Arow + kb);   // A fragment (16x4 fp32)
    #pragma unroll
    for (int j = 0; j < 8; ++j) {            // 8 N-tiles -> 128 output cols
      const v2f b = *(const v2f*)&ldsW[(16 * j + ln) * pitch + kb + kofs];
      acc[j] = WMMA_F32(a, b, acc[j]);
    }
  }

  const int rowb = r0 + 8 * lh;
  #pragma unroll
  for (int j = 0; j < 8; ++j) {
    const int col = col0 + 16 * j + ln;
    const float bv = bias1[col] + bias2[col];
    #pragma unroll
    for (int r = 0; r < 8; ++r)
      Y[(size_t)(rowb + r) * G4 + col] = acc[j][r] + bv;
  }
}

// ---------------------------------------------------------------------------
// lstm_rec: per-workgroup private recurrence over a 16-row batch tile.
// 32 WGs x 256 thr (8 waves). Per step: z = xz[t] + h . w_hh^T (16x512x128),
// wave w owns N-tiles {w, w+8, w+16, w+24} == columns 16w..16w+15 of gates
// i,f,g,o -> the cell state slice stays in this wave's registers (D-layout).
// LDS: w_hh [512][132] (270KB, CDNA5 320KB WGP LDS) + h [16][132].
// xz[t] is register software-pipelined one step ahead (loads for t+1 issue
// before the K-loop of step t and are consumed at step t+1).
// ---------------------------------------------------------------------------
__global__ __launch_bounds__(256) void lstm_rec(
    const float* __restrict__ xz, const float* __restrict__ w_hh,
    float* __restrict__ out_all, float* __restrict__ out_last, int nT)
{
  extern __shared__ float lds[];
  float* ldsW = lds;               // [512][132]
  float* ldsH = lds + 512 * 132;   // [16][132]
  const int tid  = threadIdx.x;
  const int wave = tid >> 5;
  const int lane = tid & 31;
  const int b0   = blockIdx.x * 16;

  // async-stage w_hh [512][128] into padded LDS (16B per lane per op)
  for (int i4 = tid; i4 < (512 * 128 / 4); i4 += 256) {
    const int jj = i4 >> 5;
    const int kq = (i4 & 31) << 2;
    async_g2l_b128((unsigned)(uintptr_t)&ldsW[jj * 132 + kq],
                   w_hh + jj * 128 + kq);
  }
  for (int i = tid; i < 16 * 132; i += 256) ldsH[i] = 0.0f;  // h0 = 0

  const int ln   = lane & 15;
  const int lh   = lane >> 4;
  const int kofs = 2 * lh;
  v8f c = (v8f)(0.0f);            // c0 = 0 (this wave's 16x16 slice, D-layout)

  // base pointer for this lane's xz elements (add g*Hn + r*G4 + t*Bsz*G4)
  const float* xzb = xz + (size_t)(b0 + 8 * lh) * G4 + 16 * wave + ln;

  // prologue: load xz[0] into the pipeline registers
  v8f nxt[4];
  #pragma unroll
  for (int g = 0; g < 4; ++g) {
    const float* p = xzb + g * Hn;
    #pragma unroll
    for (int r = 0; r < 8; ++r) nxt[g][r] = p[(size_t)r * G4];
  }

  wait_async0();                  // w_hh staged (own wave's copies)

  for (int t = 0; t < nT; ++t) {
    __syncthreads();              // h(t-1) + staged weights visible

    // consume pipelined xz[t], immediately issue loads for xz[t+1]
    v8f acc[4] = {nxt[0], nxt[1], nxt[2], nxt[3]};
    if (t + 1 < nT) {
      const float* pt = xzb + (size_t)(t + 1) * Bsz * G4;
      #pragma unroll
      for (int g = 0; g < 4; ++g) {
        const float* p = pt + g * Hn;
        #pragma unroll
        for (int r = 0; r < 8; ++r) nxt[g][r] = p[(size_t)r * G4];
        if (t + 2 < nT) __builtin_prefetch(p + (size_t)Bsz * G4, 0, 0);
      }
    }

    // z += h . w_hh^T : K=128 in 32 WMMA k-steps, 4 gate tiles per wave
    for (int kb = 0; kb < Hn; kb += 4) {
      const v2f a = *(const v2f*)&ldsH[ln * 132 + kb + kofs];
      #pragma unroll
      for (int g = 0; g < 4; ++g) {
        const v2f b = *(const v2f*)&ldsW[(g * Hn + 16 * wave + ln) * 132 + kb + kofs];
        acc[g] = WMMA_F32(a, b, acc[g]);
      }
    }

    __syncthreads();              // everyone done reading h(t-1)

    // gates -> cell update -> h(t)
    #pragma unroll
    for (int r = 0; r < 8; ++r) {
      const float iv = sig_(acc[0][r]);
      const float fv = sig_(acc[1][r]);
      const float gv = tanh_(acc[2][r]);
      const float ov = sig_(acc[3][r]);
      const float cv = fv * c[r] + iv * gv;
      c[r] = cv;
      const float hv = ov * tanh_(cv);
      const int row = r + 8 * lh;            // batch row in tile
      const int col = 16 * wave + ln;        // hidden index
      ldsH[row * 132 + col] = hv;
      if (out_all)
        out_all[((size_t)t * Bsz + b0 + row) * Hn + col] = hv;
      else if (t == nT - 1)
        out_last[(size_t)(b0 + row) * Hn + col] = hv;
    }
  }
}

// ---------------------------------------------------------------------------
// fc_head: out[b] = wr . relu(W2 . relu(W1 . h[b] + b1) + b2) + br  (tiny)
// ---------------------------------------------------------------------------
__global__ __launch_bounds__(128) void fc_head(
    const float* __restrict__ hlast,
    const float* __restrict__ w1, const float* __restrict__ b1,
    const float* __restrict__ w2, const float* __restrict__ b2,
    const float* __restrict__ wr, const float* __restrict__ br,
    float* __restrict__ out)
{
  __shared__ float t1[128 * 64];
  __shared__ float t2[128 * 32];
  const int tid = threadIdx.x;
  const int b   = blockIdx.x * 128 + tid;
  const float* h = hlast + (size_t)b * Hn;

  for (int j = 0; j < 64; ++j) {
    float s = b1[j];
    const float* w = w1 + j * Hn;
    for (int k = 0; k < Hn; ++k) s += w[k] * h[k];
    t1[tid * 64 + j] = s > 0.0f ? s : 0.0f;
  }
  for (int j = 0; j < 32; ++j) {
    float s = b2[j];
    const float* w = w2 + j * 64;
    for (int k = 0; k < 64; ++k) s += w[k] * t1[tid * 64 + k];
    t2[tid * 32 + j] = s > 0.0f ? s : 0.0f;
  }
  float s = br[0];
  for (int k = 0; k < 32; ++k) s += wr[k] * t2[tid * 32 + k];
  out[b] = s;
}

// ---------------------------------------------------------------------------
extern "C" void kernel_launch(void* const* d_in, const int* in_sizes, int n_in,
                              void* d_out, int out_size, void* d_ws, size_t ws_size,
                              hipStream_t stream) {
  const float* x     = (const float*)d_in[0];
  const float* w_ih0 = (const float*)d_in[1];
  const float* w_hh0 = (const float*)d_in[2];
  const float* b_ih0 = (const float*)d_in[3];
  const float* b_hh0 = (const float*)d_in[4];
  const float* w_ih1 = (const float*)d_in[5];
  const float* w_hh1 = (const float*)d_in[6];
  const float* b_ih1 = (const float*)d_in[7];
  const float* b_hh1 = (const float*)d_in[8];
  const float* w_fc1 = (const float*)d_in[9];
  const float* b_fc1 = (const float*)d_in[10];
  const float* w_fc2 = (const float*)d_in[11];
  const float* b_fc2 = (const float*)d_in[12];
  const float* w_reg = (const float*)d_in[13];
  const float* b_reg = (const float*)d_in[14];
  float* out = (float*)d_out;

  float* ws    = (float*)d_ws;
  float* xz    = ws;                                  // [T*B][512]  (512 MB, reused)
  float* out0  = ws   + (size_t)Tn * Bsz * G4;        // [T][B][H]   (128 MB)
  float* hlast = out0 + (size_t)Tn * Bsz * Hn;        // [B][H]

  const int ldsG1  = 128 * 68  * (int)sizeof(float);  // K=64  slice, pitch 68
  const int ldsG2  = 128 * 132 * (int)sizeof(float);  // K=128 slice, pitch 132
  const int ldsRec = (512 * 132 + 16 * 132) * (int)sizeof(float);  // ~272 KB

  (void)hipFuncSetAttribute((const void*)xz_gemm,
                            hipFuncAttributeMaxDynamicSharedMemorySize, ldsG2);
  (void)hipFuncSetAttribute((const void*)lstm_rec,
                            hipFuncAttributeMaxDynamicSharedMemorySize, ldsRec);

  const dim3 blk(256);

  // Layer 0: input GEMM, then private-per-batch-tile recurrence
  xz_gemm<<<dim3(2048, 4), blk, ldsG1, stream>>>(
      x, Tn * Dn, Dn, w_ih0, 64, 6, 68, b_ih0, b_hh0, xz);
  lstm_rec<<<dim3(32), blk, ldsRec, stream>>>(xz, w_hh0, out0, nullptr, Tn);

  // Layer 1: input GEMM (reuse xz scratch), recurrence emitting only h_T
  xz_gemm<<<dim3(2048, 4), blk, ldsG2, stream>>>(
      out0, Hn, Bsz * Hn, w_ih1, 128, 7, 132, b_ih1, b_hh1, xz);
  lstm_rec<<<dim3(32), blk, ldsRec, stream>>>(xz, w_hh1, nullptr, hlast, Tn);

  // Head
  fc_head<<<dim3(4), dim3(128), 0, stream>>>(
      hlast, w_fc1, b_fc1, w_fc2, b_fc2, w_reg, b_reg, out);
}